// GCN_3702261809599
// MI455X (gfx1250) — compile-verified
//
#include <hip/hip_runtime.h>
#include <hip/hip_bf16.h>
#include <math.h>

// ---------------- problem constants (fixed by the reference) ----------------
#define GN    50000   // nodes
#define GE    800000  // edges
#define F_IN  256
#define HDIM  96
#define CDIM  40
#define NLAYERS 3

typedef __attribute__((ext_vector_type(2))) float v2f;
typedef __attribute__((ext_vector_type(8))) float v8f;

// ---------------------------------------------------------------------------
// Degree / normalization kernels
// ---------------------------------------------------------------------------
__global__ void gcn_init_deg(float* deg) {
    int i = blockIdx.x * blockDim.x + threadIdx.x;
    if (i < GN) deg[i] = 1.0f;               // self loop contributes 1
}

__global__ void gcn_accum_deg(const long long* __restrict__ dst, float* deg) {
    int e = blockIdx.x * blockDim.x + threadIdx.x;
    if (e < GE) atomicAdd(&deg[(int)dst[e]], 1.0f);
}

__global__ void gcn_finalize_dinv(float* deg) {
    int i = blockIdx.x * blockDim.x + threadIdx.x;
    if (i < GN) deg[i] = rsqrtf(deg[i]);     // deg >= 1 always (self loops)
}

// ---------------------------------------------------------------------------
// Generic fp32 WMMA GEMM:  C[M x P] = A[M x K] @ W[K x P]  (+bias, +relu)
//   M = 50000 (divisible by 16), K in {256, 96} (divisible by 4), P in {96, 40}
//   One wave computes one 16x16 output tile via V_WMMA_F32_16X16X4_F32.
//   5 waves / block, 625 blocks -> exactly 3125 row tiles (no EXEC masking).
// ---------------------------------------------------------------------------
#define GEMM_WAVES 5
#define GEMM_THREADS (GEMM_WAVES * 32)

__global__ __launch_bounds__(GEMM_THREADS)
void gcn_gemm_wmma_f32(const float* __restrict__ A, const float* __restrict__ W,
                       const float* __restrict__ bias, float* __restrict__ Cout,
                       int K, int P, int applyRelu)
{
    const int lane = threadIdx.x & 31;
    const int wave = threadIdx.x >> 5;
    const int col0 = blockIdx.y * 16;
    const int rowTile = blockIdx.x * GEMM_WAVES + wave;   // < 3125 by construction

    // Stage the [K x 16] weight panel in LDS, zero-padded past P.
    __shared__ float ldsW[256 * 16];                       // 16 KB max (K<=256)
    for (int idx = threadIdx.x; idx < K * 16; idx += GEMM_THREADS) {
        int k = idx >> 4;
        int n = idx & 15;
        int c = col0 + n;
        ldsW[idx] = (c < P) ? W[k * P + c] : 0.0f;
    }
    __syncthreads();

    const int m    = lane & 15;      // A: row within tile; B/C: column within tile
    const int half = lane >> 4;      // selects K pair {0,1} vs {2,3}
    const float* arow = A + (size_t)(rowTile * 16 + m) * (size_t)K;

    v8f acc = {};
    for (int k0 = 0; k0 < K; k0 += 4) {
        const int ka = k0 + half * 2;
        v2f a, b;
        a[0] = arow[ka];
        a[1] = arow[ka + 1];
        b[0] = ldsW[ka * 16 + m];
        b[1] = ldsW[(ka + 1) * 16 + m];
        // D = A(16x4) * B(4x16) + C ; fp32 matrix pipe, EXEC all ones.
        acc = __builtin_amdgcn_wmma_f32_16x16x4_f32(
            /*neg_a=*/false, a, /*neg_b=*/false, b,
            /*c_mod=*/(short)0, acc, /*reuse_a=*/false, /*reuse_b=*/false);
    }

    const int c = col0 + m;
    if (c < P) {
        const float bv = bias ? bias[c] : 0.0f;
#pragma unroll
        for (int v = 0; v < 8; ++v) {
            int r = rowTile * 16 + v + 8 * half;
            float val = acc[v] + bv;
            if (applyRelu) val = fmaxf(val, 0.0f);
            Cout[(size_t)r * (size_t)P + c] = val;
        }
    }
}

// ---------------------------------------------------------------------------
// Aggregation: agg initialized with bias, then atomically accumulate
//   agg[dst] += t[src] * dinv[src]*dinv[dst]   over E edges + N self loops.
// One wave per edge; lanes cover the 96-float feature row (3 per lane).
// ---------------------------------------------------------------------------
__global__ void gcn_init_bias(float* __restrict__ agg, const float* __restrict__ bias) {
    int i = blockIdx.x * blockDim.x + threadIdx.x;
    if (i < GN * HDIM) agg[i] = bias[i % HDIM];
}

__global__ void gcn_scatter(const float* __restrict__ t,
                            const long long* __restrict__ src,
                            const long long* __restrict__ dst,
                            const float* __restrict__ dinv,
                            float* __restrict__ agg)
{
    const int lane   = threadIdx.x & 31;
    const int warpId = (blockIdx.x * blockDim.x + threadIdx.x) >> 5;
    const int nwarps = (gridDim.x * blockDim.x) >> 5;
    const int total  = GE + GN;
    for (int e = warpId; e < total; e += nwarps) {
        int s, d;
        if (e < GE) { s = (int)src[e]; d = (int)dst[e]; }
        else        { s = e - GE; d = s; }
        const float nm = dinv[s] * dinv[d];
        const float* hs = t   + (size_t)s * HDIM;
        float*       od = agg + (size_t)d * HDIM;
#pragma unroll
        for (int j = 0; j < 3; ++j) {
            int f = lane + 32 * j;                 // HDIM = 96 = 3 * 32
            atomicAdd(od + f, hs[f] * nm);
        }
    }
}

__global__ void gcn_relu(float* __restrict__ h, int n) {
    int i = blockIdx.x * blockDim.x + threadIdx.x;
    if (i < n) h[i] = fmaxf(h[i], 0.0f);
}

// ---------------------------------------------------------------------------
// Row-wise log_softmax over [GN x CDIM], one wave per row (wave32).
// ---------------------------------------------------------------------------
__global__ void gcn_log_softmax(float* __restrict__ logits) {
    const int lane   = threadIdx.x & 31;
    const int warpId = (blockIdx.x * blockDim.x + threadIdx.x) >> 5;
    if (warpId >= GN) return;
    float* row = logits + (size_t)warpId * CDIM;

    const bool hi = (lane + 32) < CDIM;            // lanes 0..7 cover cols 32..39
    float v0 = row[lane];                          // lane < 32 <= CDIM-? (CDIM=40>32, ok)
    float v1 = hi ? row[lane + 32] : -INFINITY;

    float mx = fmaxf(v0, v1);
#pragma unroll
    for (int off = 16; off > 0; off >>= 1) mx = fmaxf(mx, __shfl_xor(mx, off, 32));

    float s = expf(v0 - mx) + (hi ? expf(v1 - mx) : 0.0f);
#pragma unroll
    for (int off = 16; off > 0; off >>= 1) s += __shfl_xor(s, off, 32);

    const float lse = mx + logf(s);
    row[lane] = v0 - lse;
    if (hi) row[lane + 32] = v1 - lse;
}

// ---------------------------------------------------------------------------
// Host-side orchestration
// ---------------------------------------------------------------------------
extern "C" void kernel_launch(void* const* d_in, const int* in_sizes, int n_in,
                              void* d_out, int out_size, void* d_ws, size_t ws_size,
                              hipStream_t stream)
{
    (void)in_sizes; (void)n_in; (void)out_size; (void)ws_size;

    const float*     x       = (const float*)d_in[0];            // [N, 256]
    const long long* eidx    = (const long long*)d_in[1];        // [2, E] int64
    const float*     w_in    = (const float*)d_in[2];            // [256, 96]
    const float*     conv_w  = (const float*)d_in[3];            // [3, 96, 96]
    const float*     conv_b  = (const float*)d_in[4];            // [3, 96]
    const float*     w_out   = (const float*)d_in[5];            // [96, 40]
    float*           out     = (float*)d_out;                    // [N, 40]

    const long long* src = eidx;            // edge_index[0]
    const long long* dst = eidx + GE;       // edge_index[1]

    // Workspace layout (floats): dinv[50176 pad] | hA[N*96] | hB[N*96]
    float* dinv = (float*)d_ws;
    float* hA   = dinv + 50176;
    float* hB   = hA + (size_t)GN * HDIM;

    const int T = 256;

    // 1) symmetric GCN normalization: dinv = rsqrt(deg with self loops)
    gcn_init_deg     <<<(GN + T - 1) / T, T, 0, stream>>>(dinv);
    gcn_accum_deg    <<<(GE + T - 1) / T, T, 0, stream>>>(dst, dinv);
    gcn_finalize_dinv<<<(GN + T - 1) / T, T, 0, stream>>>(dinv);

    // 2) input transform: hA = x @ w_in          [N,256]x[256,96]
    dim3 gemmGrid(625, HDIM / 16);          // 3125 row tiles / 5 waves, 6 col tiles
    gcn_gemm_wmma_f32<<<gemmGrid, GEMM_THREADS, 0, stream>>>(
        x, w_in, nullptr, hA, F_IN, HDIM, 0);

    // 3) GCN layers: t = h@W; agg(bias) += t[src]*norm -> dst; relu
    const int NH = GN * HDIM;
    const int scatterBlocks = (GE + GN + 7) / 8;   // 8 waves/block, 1 wave/edge
    for (int l = 0; l < NLAYERS; ++l) {
        gcn_gemm_wmma_f32<<<gemmGrid, GEMM_THREADS, 0, stream>>>(
            hA, conv_w + (size_t)l * HDIM * HDIM, nullptr, hB, HDIM, HDIM, 0);
        gcn_init_bias<<<(NH + T - 1) / T, T, 0, stream>>>(hA, conv_b + (size_t)l * HDIM);
        gcn_scatter  <<<scatterBlocks, T, 0, stream>>>(hB, src, dst, dinv, hA);
        gcn_relu     <<<(NH + T - 1) / T, T, 0, stream>>>(hA, NH);
    }

    // 4) output transform: out = hA @ w_out      [N,96]x[96,40] (3 col tiles)
    dim3 outGrid(625, (CDIM + 15) / 16);
    gcn_gemm_wmma_f32<<<outGrid, GEMM_THREADS, 0, stream>>>(
        hA, w_out, nullptr, out, HDIM, CDIM, 0);

    // 5) row-wise log_softmax, one wave per node
    gcn_log_softmax<<<(GN * 32 + T - 1) / T, T, 0, stream>>>(out);
}